// GIN_23210003268004
// MI455X (gfx1250) — compile-verified
//
#include <hip/hip_runtime.h>
#include <hip/hip_bf16.h>

// ---------------------------------------------------------------------------
// GIN layer on MI455X (gfx1250, wave32):
//   agg = segment_sum(x[src], dst)                    -> edge_scatter kernel
//   h2  = relu((x+agg)@W1+b1)@W2+b2 ; pooled += h2    -> fused WMMA MLP kernel
//   out = pooled@Wl + bl                              -> final kernel
// fp32 end to end; matrix ops via V_WMMA_F32_16X16X4_F32.
// ---------------------------------------------------------------------------

#define N_NODES  100000
#define N_EDGES  1200000
#define IN_DIM   64
#define HIDDEN   128
#define N_GRAPHS 512

typedef __attribute__((ext_vector_type(2))) float v2f;
typedef __attribute__((ext_vector_type(8))) float v8f;

// ---------------------------------------------------------------- zero fill
__global__ void gin_zero_f4(float* __restrict__ p, int n4) {
    int i = blockIdx.x * blockDim.x + threadIdx.x;
    if (i < n4) {
        float4 z; z.x = 0.f; z.y = 0.f; z.z = 0.f; z.w = 0.f;
        ((float4*)p)[i] = z;
    }
}

// ------------------------------------------------------------- edge scatter
// 16 lanes per edge, each lane moves one float4 chunk of the 64-float row.
// x and agg are L2-resident (25.6 MB each on a 192 MB L2), so the gathered
// reads and GLOBAL_ATOMIC_ADD_F32 traffic stay on-chip.
__global__ void gin_edge_scatter(const float* __restrict__ x,
                                 const int*   __restrict__ src,
                                 const int*   __restrict__ dst,
                                 float*       __restrict__ agg) {
    int t = blockIdx.x * blockDim.x + threadIdx.x;
    int e = t >> 4;
    if (e >= N_EDGES) return;
    int chunk = (t & 15) << 2;
    int s = src[e];
    int d = dst[e];
    const float4 v = *(const float4*)(x + (size_t)s * IN_DIM + chunk);
    float* o = agg + (size_t)d * IN_DIM + chunk;
    atomicAdd(o + 0, v.x);
    atomicAdd(o + 1, v.y);
    atomicAdd(o + 2, v.z);
    atomicAdd(o + 3, v.w);
}

// ---------------------------------------------------- fused MLP + pooling
// Block = 256 threads (8 waves), owns 16 node rows.
// Wave w computes the 16x16 tile for hidden columns [16w, 16w+16).
// GEMM K stepped by 4 via v_wmma_f32_16x16x4_f32.
__global__ void __launch_bounds__(256)
gin_mlp_pool(const float* __restrict__ x,
             const float* __restrict__ agg,
             const float* __restrict__ W1, const float* __restrict__ b1,
             const float* __restrict__ W2, const float* __restrict__ b2,
             const int*   __restrict__ batch,
             float*       __restrict__ pooled) {
    // +4 float row pad: LDS bank = (row*4 + k) % 64 -> 16 rows hit 16 banks.
    __shared__ float sh_h [16 * (IN_DIM + 4)];   // h = x + agg   (16 x 64)
    __shared__ float sh_l1[16 * (HIDDEN + 4)];   // relu(h@W1+b1) (16 x 128)
    __shared__ int   sh_g [16];

    const int t    = threadIdx.x;
    const int row0 = blockIdx.x * 16;            // 6250 * 16 == 100000 exactly

    // ---- phase 1: h tile -> LDS (256 threads x 1 float4 = 1024 floats)
    {
        int fidx = t << 2;
        int row  = fidx >> 6;
        int col  = fidx & 63;
        size_t g = (size_t)(row0 + row) * IN_DIM + col;
        float4 xv = *(const float4*)(x + g);
        float4 av = *(const float4*)(agg + g);
        float4 hv;
        hv.x = xv.x + av.x; hv.y = xv.y + av.y;
        hv.z = xv.z + av.z; hv.w = xv.w + av.w;
        *(float4*)(&sh_h[row * (IN_DIM + 4) + col]) = hv;
        if (t < 16) sh_g[t] = batch[row0 + t];
    }
    __syncthreads();

    const int wave  = t >> 5;
    const int lane  = t & 31;
    const int mrow  = lane & 15;          // A row / B-C column within tile
    const int koff  = (lane >> 4) << 1;   // 0 for lanes 0-15, 2 for 16-31
    const int n0    = wave << 4;
    const int ncol  = n0 + mrow;          // hidden column this lane owns in C
    const int mbase = (lane >> 4) << 3;   // C rows: 0..7 or 8..15

    // ---- phase 2: layer1 = relu(h @ W1 + b1), K = 64
    v8f c;
    {
        float bias = b1[ncol];
        #pragma unroll
        for (int r = 0; r < 8; ++r) c[r] = bias;

        const float* aP = &sh_h[mrow * (IN_DIM + 4) + koff];
        const float* bP = &W1[(size_t)koff * HIDDEN + ncol];
        #pragma unroll
        for (int k0 = 0; k0 < IN_DIM; k0 += 4) {
            v2f a, b;
            a.x = aP[k0];
            a.y = aP[k0 + 1];
            b.x = bP[(size_t)k0 * HIDDEN];
            b.y = bP[(size_t)(k0 + 1) * HIDDEN];
            c = __builtin_amdgcn_wmma_f32_16x16x4_f32(
                    false, a, false, b, (short)0, c, false, false);
        }
        #pragma unroll
        for (int r = 0; r < 8; ++r)
            sh_l1[(mbase + r) * (HIDDEN + 4) + ncol] = fmaxf(c[r], 0.0f);
    }
    __syncthreads();

    // ---- phase 3: h2 = layer1 @ W2 + b2, K = 128
    v8f c2;
    {
        float bias = b2[ncol];
        #pragma unroll
        for (int r = 0; r < 8; ++r) c2[r] = bias;

        const float* aP = &sh_l1[mrow * (HIDDEN + 4) + koff];
        const float* bP = &W2[(size_t)koff * HIDDEN + ncol];
        #pragma unroll
        for (int k0 = 0; k0 < HIDDEN; k0 += 4) {
            v2f a, b;
            a.x = aP[k0];
            a.y = aP[k0 + 1];
            b.x = bP[(size_t)k0 * HIDDEN];
            b.y = bP[(size_t)(k0 + 1) * HIDDEN];
            c2 = __builtin_amdgcn_wmma_f32_16x16x4_f32(
                    false, a, false, b, (short)0, c2, false, false);
        }
    }

    // ---- phase 4: global_add_pool — atomics into L2-resident pooled[512,128]
    #pragma unroll
    for (int r = 0; r < 8; ++r) {
        int m = mbase + r;
        int g = sh_g[m];
        atomicAdd(&pooled[(size_t)g * HIDDEN + ncol], c2[r]);
    }
}

// ------------------------------------------------------------- final linear
__global__ void gin_final(const float* __restrict__ pooled,
                          const float* __restrict__ Wl,
                          const float* __restrict__ bl,
                          float*       __restrict__ out) {
    int g = blockIdx.x * blockDim.x + threadIdx.x;
    if (g >= N_GRAPHS) return;
    const float* p = pooled + (size_t)g * HIDDEN;
    float acc = bl[0];
    #pragma unroll 8
    for (int k = 0; k < HIDDEN; ++k) acc += p[k] * Wl[k];
    out[g] = acc;
}

// ---------------------------------------------------------------------------
extern "C" void kernel_launch(void* const* d_in, const int* in_sizes, int n_in,
                              void* d_out, int out_size, void* d_ws, size_t ws_size,
                              hipStream_t stream) {
    const float* x     = (const float*)d_in[0];
    const int*   eidx  = (const int*)  d_in[1];   // [2, N_EDGES]: src then dst
    const int*   batch = (const int*)  d_in[2];
    const float* W1    = (const float*)d_in[3];
    const float* b1    = (const float*)d_in[4];
    const float* W2    = (const float*)d_in[5];
    const float* b2    = (const float*)d_in[6];
    const float* Wl    = (const float*)d_in[7];
    const float* bl    = (const float*)d_in[8];
    float* outp = (float*)d_out;

    const int* src = eidx;
    const int* dst = eidx + N_EDGES;

    // workspace: agg [N_NODES, IN_DIM] then pooled [N_GRAPHS, HIDDEN]
    float* agg    = (float*)d_ws;
    float* pooled = agg + (size_t)N_NODES * IN_DIM;

    // zero agg + pooled (fresh each call; harness does not re-poison)
    {
        int n4 = (N_NODES * IN_DIM) / 4;          // 1,600,000
        gin_zero_f4<<<(n4 + 255) / 256, 256, 0, stream>>>(agg, n4);
        int p4 = (N_GRAPHS * HIDDEN) / 4;         // 16,384
        gin_zero_f4<<<(p4 + 255) / 256, 256, 0, stream>>>(pooled, p4);
    }

    // edge scatter: 16 threads per edge
    {
        long long threads = (long long)N_EDGES * 16;
        int blocks = (int)((threads + 255) / 256);
        gin_edge_scatter<<<blocks, 256, 0, stream>>>(x, src, dst, agg);
    }

    // fused MLP + pooling: 16 rows per block, exact cover (100000 = 16*6250)
    gin_mlp_pool<<<N_NODES / 16, 256, 0, stream>>>(x, agg, W1, b1, W2, b2,
                                                   batch, pooled);

    // final linear
    gin_final<<<(N_GRAPHS + 255) / 256, 256, 0, stream>>>(pooled, Wl, bl, outp);
}